// Autoencoder_41420664602992
// MI455X (gfx1250) — compile-verified
//
#include <hip/hip_runtime.h>

typedef __attribute__((ext_vector_type(2))) float v2f;
typedef __attribute__((ext_vector_type(8))) float v8f;

// Intra-wave LDS ordering fence: waits for outstanding DS ops and stops the
// compiler from reordering LDS traffic across it. All tile sharing is within
// one wave32, and CDNA5 keeps same-wave LDS ops in order, so no s_barrier needed.
#define LDS_FENCE() asm volatile("s_wait_dscnt 0" ::: "memory")

#define LA 8
#define LG 16
#define NUMF 14
#define XSTRIDE 21   // 16x20 tile, stride 21 floats: gcd(21,64)=1 -> conflict-free
#define NCHUNKS 19   // total K-chunks across all 7 GEMM tiles
#define NBIAS 7      // bias slots (6 layers + second N-tile of last layer)

// ---------------------------------------------------------------------------
// Prep kernel: pack weights+biases into per-lane WMMA B-operand layout in d_ws.
// Chunk q, lane L -> v2f { B[k][n], B[k+1][n] } with k = 4*ci + 2*(L>>4),
// n = L&15 (+colOff for the second N-tile), zero-padded out of range.
// Runs once per launch with 32 threads; cost is nil.
// ---------------------------------------------------------------------------
__global__ void pack_weights_kernel(const float* __restrict__ ew0,
                                    const float* __restrict__ ew1,
                                    const float* __restrict__ ew2,
                                    const float* __restrict__ eb0,
                                    const float* __restrict__ eb1,
                                    const float* __restrict__ eb2,
                                    const float* __restrict__ dw0,
                                    const float* __restrict__ dw1,
                                    const float* __restrict__ dw2,
                                    const float* __restrict__ db0,
                                    const float* __restrict__ db1,
                                    const float* __restrict__ db2,
                                    float* __restrict__ ws) {
  int lane = threadIdx.x;
  if (lane >= 32) return;
  int n = lane & 15;
  int h = lane >> 4;
  int q = 0;
  auto pack = [&](const float* W, int IN, int OUT, int colOff, int NC) {
    for (int ci = 0; ci < NC; ++ci, ++q) {
      int k = ci * 4 + 2 * h;
      int col = colOff + n;
      float v0 = (col < OUT && k     < IN) ? W[col * IN + k]     : 0.0f;
      float v1 = (col < OUT && k + 1 < IN) ? W[col * IN + k + 1] : 0.0f;
      ws[(q * 32 + lane) * 2]     = v0;
      ws[(q * 32 + lane) * 2 + 1] = v1;
    }
  };
  pack(ew0, 20, 10, 0, 5);   // chunks 0..4
  pack(ew1, 10,  6, 0, 3);   // 5..7
  pack(ew2,  6,  3, 0, 2);   // 8..9
  pack(dw0,  3,  6, 0, 1);   // 10
  pack(dw1,  6, 10, 0, 2);   // 11..12
  pack(dw2, 10, 20, 0, 3);   // 13..15 (output cols 0..15)
  pack(dw2, 10, 20, 16, 3);  // 16..18 (output cols 16..19)

  float* bs = ws + NCHUNKS * 64;
  auto packb = [&](const float* Bp, int OUT, int colOff, int slot) {
    int col = colOff + n;
    bs[slot * 32 + lane] = (col < OUT) ? Bp[col] : 0.0f;
  };
  packb(eb0, 10, 0, 0);
  packb(eb1,  6, 0, 1);
  packb(eb2,  3, 0, 2);
  packb(db0,  6, 0, 3);
  packb(db1, 10, 0, 4);
  packb(db2, 20, 0, 5);
  packb(db2, 20, 16, 6);
}

// ---------------------------------------------------------------------------
// One MLP layer on a 16-row LDS tile: relu(X @ W^T + b) with fp32 WMMA.
// Weights come pre-packed in registers (wv[Q0..Q0+NC-1]); A-operand reads are
// unconditional because padded LDS columns are exact zeros.
// ---------------------------------------------------------------------------
template<int Q0, int NC>
__device__ __forceinline__ void mlp_layer(float* __restrict__ X,
                                          const v2f* __restrict__ wv,
                                          float bv, int h, int n) {
  v8f c = {bv, bv, bv, bv, bv, bv, bv, bv};
#pragma unroll
  for (int ci = 0; ci < NC; ++ci) {
    int k = ci * 4 + 2 * h;
    v2f a;
    a.x = X[n * XSTRIDE + k];
    a.y = X[n * XSTRIDE + k + 1];
    c = __builtin_amdgcn_wmma_f32_16x16x4_f32(false, a, false, wv[Q0 + ci],
                                              (short)0, c, false, false);
  }
  LDS_FENCE();   // all lanes done reading X before overwriting it
#pragma unroll
  for (int j = 0; j < 8; ++j)
    X[(h * 8 + j) * XSTRIDE + n] = fmaxf(c[j], 0.0f);
  LDS_FENCE();
}

__global__ void __launch_bounds__(256)
autoenc_wmma_kernel(const int* __restrict__ artist_idx,
                    const int* __restrict__ artist_len,
                    const int* __restrict__ genre_idx,
                    const int* __restrict__ genre_len,
                    const float* __restrict__ numeric,
                    const float* __restrict__ artist_emb,
                    const float* __restrict__ genre_emb,
                    const float* __restrict__ wpack,
                    float* __restrict__ out, int nrows) {
  __shared__ float Xs[8][16 * XSTRIDE];         // one 16x20 tile per wave
  const int lane = threadIdx.x & 31;
  const int wslot = threadIdx.x >> 5;
  float* X = Xs[wslot];
  const int n = lane & 15;                      // column index (B/C/D) == A row
  const int h = lane >> 4;                      // lane half
  const int r0 = (blockIdx.x * 8 + wslot) * 16; // first row of this wave's tile

  // ---- Preload packed weights/biases: branchless, L1-resident, overlaps gather
  v2f wv[NCHUNKS];
#pragma unroll
  for (int q = 0; q < NCHUNKS; ++q)
    wv[q] = *reinterpret_cast<const v2f*>(wpack + (q * 32 + lane) * 2);
  float bvs[NBIAS];
#pragma unroll
  for (int l = 0; l < NBIAS; ++l)
    bvs[l] = wpack[NCHUNKS * 64 + l * 32 + lane];

  // ---- Phase 0: build x = [artist_mean(3) | numeric(14) | genre_mean(3)] in LDS
  if (lane < 16) {
    int r = r0 + lane;
    if (r < nrows) {
      float* xr = X + lane * XSTRIDE;

      int alen = artist_len[r];
      float a0 = 0.f, a1 = 0.f, a2 = 0.f;
#pragma unroll
      for (int j = 0; j < LA; ++j) {
        if (j < alen) {
          int id = artist_idx[r * LA + j];
          const float* e = artist_emb + id * 3;   // 384 KB table: L2-resident
          a0 += e[0]; a1 += e[1]; a2 += e[2];
        }
      }
      float ainv = 1.0f / (float)alen;
      xr[0] = a0 * ainv; xr[1] = a1 * ainv; xr[2] = a2 * ainv;

#pragma unroll
      for (int q = 0; q < NUMF; ++q)
        xr[3 + q] = numeric[(size_t)r * NUMF + q];

      int glen = genre_len[r];
      float g0 = 0.f, g1 = 0.f, g2 = 0.f;
#pragma unroll
      for (int j = 0; j < LG; ++j) {
        if (j < glen) {
          int id = genre_idx[r * LG + j];
          const float* e = genre_emb + id * 3;    // 32 KB table: cache-resident
          g0 += e[0]; g1 += e[1]; g2 += e[2];
        }
      }
      float ginv = 1.0f / (float)glen;
      xr[17] = g0 * ginv; xr[18] = g1 * ginv; xr[19] = g2 * ginv;
    }
  }
  LDS_FENCE();

  // ---- Encoder + decoder (all but last layer): pure ds_load + v_wmma
  mlp_layer< 0, 5>(X, wv, bvs[0], h, n);   // 20 -> 10
  mlp_layer< 5, 3>(X, wv, bvs[1], h, n);   // 10 ->  6
  mlp_layer< 8, 2>(X, wv, bvs[2], h, n);   //  6 ->  3
  mlp_layer<10, 1>(X, wv, bvs[3], h, n);   //  3 ->  6
  mlp_layer<11, 2>(X, wv, bvs[4], h, n);   //  6 -> 10

  // ---- Final layer 10 -> 20: two N-tiles, straight to HBM with ReLU
  v8f c0 = {bvs[5], bvs[5], bvs[5], bvs[5], bvs[5], bvs[5], bvs[5], bvs[5]};
  v8f c1 = {bvs[6], bvs[6], bvs[6], bvs[6], bvs[6], bvs[6], bvs[6], bvs[6]};
#pragma unroll
  for (int ci = 0; ci < 3; ++ci) {
    int k = ci * 4 + 2 * h;
    v2f a;
    a.x = X[n * XSTRIDE + k];
    a.y = X[n * XSTRIDE + k + 1];
    c0 = __builtin_amdgcn_wmma_f32_16x16x4_f32(false, a, false, wv[13 + ci],
                                               (short)0, c0, false, false);
    c1 = __builtin_amdgcn_wmma_f32_16x16x4_f32(false, a, false, wv[16 + ci],
                                               (short)0, c1, false, false);
  }

  if (r0 + 16 <= nrows) {               // wave-uniform fast path: no per-row guards
#pragma unroll
    for (int j = 0; j < 8; ++j) {
      size_t row = (size_t)(r0 + h * 8 + j);
      out[row * 20 + n] = fmaxf(c0[j], 0.0f);
      if (n < 4)
        out[row * 20 + 16 + n] = fmaxf(c1[j], 0.0f);
    }
  } else {                              // ragged tail (unused when B % 16 == 0)
#pragma unroll
    for (int j = 0; j < 8; ++j) {
      int row = r0 + h * 8 + j;
      if (row < nrows) {
        out[(size_t)row * 20 + n] = fmaxf(c0[j], 0.0f);
        if (n < 4)
          out[(size_t)row * 20 + 16 + n] = fmaxf(c1[j], 0.0f);
      }
    }
  }
}

extern "C" void kernel_launch(void* const* d_in, const int* in_sizes, int n_in,
                              void* d_out, int out_size, void* d_ws, size_t ws_size,
                              hipStream_t stream) {
  // setup_inputs() dict order, lists flattened leaf-by-leaf:
  // 0 artist_idx [B*8] i32, 1 artist_len [B] i32, 2 genre_idx [B*16] i32,
  // 3 genre_len [B] i32, 4 numeric [B*14] f32, 5 artist_emb [32000*3] f32,
  // 6 genre_emb [2664*3] f32,
  // 7..9  enc_ws: [10,20],[6,10],[3,6]   10..12 enc_bs: [10],[6],[3]
  // 13..15 dec_ws: [6,3],[10,6],[20,10]  16..18 dec_bs: [6],[10],[20]
  const int*   artist_idx = (const int*)  d_in[0];
  const int*   artist_len = (const int*)  d_in[1];
  const int*   genre_idx  = (const int*)  d_in[2];
  const int*   genre_len  = (const int*)  d_in[3];
  const float* numeric    = (const float*)d_in[4];
  const float* artist_emb = (const float*)d_in[5];
  const float* genre_emb  = (const float*)d_in[6];
  float* ws = (float*)d_ws;   // needs 19*64 + 7*32 = 1440 floats = 5.8 KB
  float* out = (float*)d_out;

  pack_weights_kernel<<<1, 32, 0, stream>>>(
      (const float*)d_in[7],  (const float*)d_in[8],  (const float*)d_in[9],
      (const float*)d_in[10], (const float*)d_in[11], (const float*)d_in[12],
      (const float*)d_in[13], (const float*)d_in[14], (const float*)d_in[15],
      (const float*)d_in[16], (const float*)d_in[17], (const float*)d_in[18],
      ws);

  int B = in_sizes[1];                 // rows
  int blocks = (B + 127) / 128;        // 8 waves/block * 16 rows/wave
  autoenc_wmma_kernel<<<blocks, 256, 0, stream>>>(
      artist_idx, artist_len, genre_idx, genre_len, numeric,
      artist_emb, genre_emb, ws, out, B);
}